// RotatE_81595788689595
// MI455X (gfx1250) — compile-verified
//
#include <hip/hip_runtime.h>
#include <hip/hip_bf16.h>
#include <math.h>

#define BATCH 256
#define N_NEG 4096
#define EMB   256
#define HALF  (EMB / 2)

typedef __attribute__((ext_vector_type(2))) float v2f;
typedef __attribute__((ext_vector_type(8))) float v8f;

// ---------------------------------------------------------------------------
// Kernel 1: complex rotation + per-row squared norm of rotated head.
// rot[b][j]      = re[j]*cos(p[j]) - im[j]*sin(p[j])        j in [0,128)
// rot[b][j+128]  = re[j]*sin(p[j]) + im[j]*cos(p[j])
// rn[b]          = sum_j (re^2 + im^2)   (rotation preserves norm)
// One block of 128 threads per batch row; deterministic LDS tree reduction.
// ---------------------------------------------------------------------------
__global__ void __launch_bounds__(128)
rotate_kernel(const float* __restrict__ head,
              const int*   __restrict__ rel_id,
              const float* __restrict__ rel_emb,
              float* __restrict__ rot,
              float* __restrict__ rn) {
    const int b = blockIdx.x;
    const int j = threadIdx.x;            // 0..127

    const float re = head[b * EMB + j];
    const float im = head[b * EMB + HALF + j];
    const int   r  = rel_id[b];
    const float ph = rel_emb[r * HALF + j];

    float s, c;
    __sincosf(ph, &s, &c);

    rot[b * EMB + j]        = re * c - im * s;
    rot[b * EMB + HALF + j] = re * s + im * c;

    __shared__ float sm[128];
    sm[j] = re * re + im * im;
    __syncthreads();
    #pragma unroll
    for (int stride = 64; stride > 0; stride >>= 1) {
        if (j < stride) sm[j] += sm[j + stride];
        __syncthreads();
    }
    if (j == 0) rn[b] = sm[0];
}

// ---------------------------------------------------------------------------
// Kernel 2: candidate squared norms. One block (256 threads) per candidate.
// ---------------------------------------------------------------------------
__global__ void __launch_bounds__(256)
candnorm_kernel(const float* __restrict__ cand,
                float* __restrict__ cn) {
    const int t   = blockIdx.x;
    const int tid = threadIdx.x;          // 0..255
    const float v = cand[t * EMB + tid];

    __shared__ float sm[256];
    sm[tid] = v * v;
    __syncthreads();
    #pragma unroll
    for (int stride = 128; stride > 0; stride >>= 1) {
        if (tid < stride) sm[tid] += sm[tid + stride];
        __syncthreads();
    }
    if (tid == 0) cn[t] = sm[0];
}

// ---------------------------------------------------------------------------
// Kernel 3: WMMA GEMM  dot = rot (256x256) * cand^T (256x4096), fused epilogue
//           out[m][n] = -sqrt(max(0, rn[m] + cn[n] - 2*dot))
//
// v_wmma_f32_16x16x4_f32 operand layout (32-bit A 16x4 / B 4x16):
//   lanes 0-15 : M(or N) = lane,    VGPR0 -> K=0, VGPR1 -> K=1
//   lanes 16-31: M(or N) = lane-16, VGPR0 -> K=2, VGPR1 -> K=3
// => per K-step of 4 each lane loads one float2 at column (k + 2*(lane>=16)).
//
// Block: 256 threads = 8 waves. blockIdx.y = M-tile (16 of them),
// blockIdx.x*8 + wave = N-tile (256 of them). All waves share A rows (L0 hit).
// ---------------------------------------------------------------------------
__global__ void __launch_bounds__(256)
score_kernel(const float* __restrict__ rot,
             const float* __restrict__ cand,
             const float* __restrict__ rn,
             const float* __restrict__ cn,
             float* __restrict__ out) {
    const int lane  = threadIdx.x & 31;
    const int wave  = threadIdx.x >> 5;
    const int mtile = blockIdx.y;                  // 0..15
    const int ntile = blockIdx.x * 8 + wave;       // 0..255

    const int r    = lane & 15;                    // row within tile
    const int kadj = (lane >> 4) << 1;             // 0 for lanes 0-15, 2 for 16-31

    const float* arow = rot  + (mtile * 16 + r) * EMB + kadj;
    const float* brow = cand + (ntile * 16 + r) * EMB + kadj;

    v8f acc = {};
    #pragma unroll
    for (int k = 0; k < EMB; k += 4) {
        v2f a = *(const v2f*)(arow + k);
        v2f b = *(const v2f*)(brow + k);
        // 8 args: (neg_a, A, neg_b, B, c_mod, C, reuse_a, reuse_b)
        acc = __builtin_amdgcn_wmma_f32_16x16x4_f32(
            false, a, false, b, (short)0, acc, false, false);
    }

    // C/D layout: VGPR j -> lanes 0-15: M=j, N=lane; lanes 16-31: M=j+8, N=lane-16
    const int col  = ntile * 16 + r;
    const float cnv = cn[col];
    #pragma unroll
    for (int j = 0; j < 8; ++j) {
        const int row = mtile * 16 + j + ((lane >> 4) << 3);
        const float d2 = rn[row] + cnv - 2.0f * acc[j];
        out[row * N_NEG + col] = -sqrtf(fmaxf(d2, 0.0f));
    }
}

// ---------------------------------------------------------------------------
extern "C" void kernel_launch(void* const* d_in, const int* in_sizes, int n_in,
                              void* d_out, int out_size, void* d_ws, size_t ws_size,
                              hipStream_t stream) {
    const float* head    = (const float*)d_in[0];   // (256, 256)
    const int*   rel_id  = (const int*)  d_in[1];   // (256,)
    const float* tail    = (const float*)d_in[2];   // (1, 4096, 256)
    const float* rel_emb = (const float*)d_in[3];   // (500, 128)
    float*       out     = (float*)d_out;           // (256, 4096)

    float* rot = (float*)d_ws;                      // 256*256 floats
    float* rn  = rot + BATCH * EMB;                 // 256 floats
    float* cn  = rn + BATCH;                        // 4096 floats

    rotate_kernel<<<BATCH, 128, 0, stream>>>(head, rel_id, rel_emb, rot, rn);
    candnorm_kernel<<<N_NEG, 256, 0, stream>>>(tail, cn);

    dim3 grid(N_NEG / (16 * 8), BATCH / 16);        // (32, 16)
    score_kernel<<<grid, 256, 0, stream>>>(rot, tail, rn, cn, out);
}